// EquivariantMessageLayer_66374424592460
// MI455X (gfx1250) — compile-verified
//
#include <hip/hip_runtime.h>
#include <hip/hip_bf16.h>
#include <stdint.h>

// ---------------------------------------------------------------------------
// EquivariantMessageLayer for MI455X (gfx1250, wave32)
//   Phase A: feat_w[N,192] = Linear(64->32) -> ScaledSiLU -> Linear(32->192)
//            via v_wmma_f32_16x16x4_f32 (full f32 precision)
//   Phase B: per-edge radial GEMM  radial_w = edge_radial @ Wr + br  (WMMA),
//            fused gather of feat_w[src] / vector_features[src],
//            combine, and atomic scatter-add into agg_scalars / agg_vectors.
//   LDS tile staging uses the Tensor Data Mover (tensor_load_to_lds) with the
//   D# pad fields generating the bank-conflict-avoiding row padding.
// ---------------------------------------------------------------------------

typedef float v2f __attribute__((ext_vector_type(2)));
typedef float v8f __attribute__((ext_vector_type(8)));
typedef unsigned int u32x4 __attribute__((ext_vector_type(4)));
typedef int i32x8 __attribute__((ext_vector_type(8)));
typedef int i32x4 __attribute__((ext_vector_type(4)));

#define Hdim 64
#define Rdim 32
#define TH3  192           // 3*H
#define INV_SQRT_3f 0.57735026918962576f
#define INV_SQRT_Hf 0.125f // 1/sqrt(64)
#define SILU_SCALEf 1.6666666666666667f

#if defined(__has_builtin)
#if __has_builtin(__builtin_amdgcn_tensor_load_to_lds) && __has_builtin(__builtin_amdgcn_s_wait_tensorcnt)
#define HAVE_TDM 1
#endif
#endif

#ifdef HAVE_TDM
// DMA a [tile_rows x row_len(dwords)] f32 tile from global into LDS, inserting
// 1 pad DWORD every 2^(pad_interval+1) DWORDs (=> row stride row_len+1 when
// pad_interval encodes row_len). Rows >= tensor_rows are zero-filled (OOB).
__device__ __forceinline__ void tdm_load_tile(uint32_t lds_off, const float* gsrc,
                                              uint32_t row_len, uint32_t tile_rows,
                                              uint32_t tensor_rows, uint32_t pad_interval) {
    uint64_t ga = (uint64_t)(uintptr_t)gsrc;
    u32x4 g0;
    g0.x = 1u;                                        // count=1 (valid descriptor)
    g0.y = lds_off;                                   // lds_addr (bytes)
    g0.z = (uint32_t)(ga & 0xFFFFFFFFu);              // global_addr[31:0]
    g0.w = (uint32_t)((ga >> 32) & 0x01FFFFFFu)       // global_addr[56:32]
           | (2u << 30);                              // type=2 ("image")
    i32x8 g1;
    g1[0] = (int)((2u << 16)                          // data_size = 4 bytes
                  | (1u << 20)                        // pad_enable
                  | (pad_interval << 22));            // pad_interval; pad_amount=0 (1 DWORD)
    g1[1] = (int)((row_len & 0xFFFFu) << 16);         // tensor_dim0[15:0]
    g1[2] = (int)(((row_len >> 16) & 0xFFFFu)         // tensor_dim0[31:16]
                  | ((tensor_rows & 0xFFFFu) << 16)); // tensor_dim1[15:0]
    g1[3] = (int)(((tensor_rows >> 16) & 0xFFFFu)     // tensor_dim1[31:16]
                  | ((row_len & 0xFFFFu) << 16));     // tile_dim0
    g1[4] = (int)(tile_rows & 0xFFFFu);               // tile_dim1; tile_dim2=0
    g1[5] = (int)row_len;                             // tensor_dim0_stride[31:0]
    g1[6] = 0;
    g1[7] = 0;
    i32x4 z4 = {0, 0, 0, 0};
    i32x8 z8 = {0, 0, 0, 0, 0, 0, 0, 0};
    // 6-arg form (clang-23 / therock-10.0 headers)
    __builtin_amdgcn_tensor_load_to_lds(g0, g1, z4, z4, z8, 0);
}
#endif

// ---------------------------------------------------------------- zero output
__global__ void emsg_zero_kernel(float* __restrict__ p, long n4) {
    long i = (long)blockIdx.x * blockDim.x + threadIdx.x;
    if (i < n4) {
        ((float4*)p)[i] = make_float4(0.f, 0.f, 0.f, 0.f);
    }
}

// ------------------------------------------------------- node feature projector
// block = 128 threads (4 waves). Each wave computes a 16-node tile; block = 64 nodes.
// GEMM1: [16x64]@[64x32] -> SiLU*scale -> GEMM2: [16x32]@[32x192] -> feat_w
__global__ void __launch_bounds__(128)
emsg_featproj_kernel(const float* __restrict__ nf,    // [N,64]
                     const float* __restrict__ Wf1,   // [64,32]
                     const float* __restrict__ bf1,   // [32]
                     const float* __restrict__ Wf2,   // [32,192]
                     const float* __restrict__ bf2,   // [192]
                     float* __restrict__ feat_w,      // [N,192]
                     int N) {
    __shared__ float sNF[64 * 65];          // 64 nodes x 64 ch, stride 65 (pad)
    __shared__ float sH[4 * 16 * 33];       // per-wave 16x32 hidden, stride 33

    const int tid  = threadIdx.x;
    const int lane = tid & 31;
    const int wv   = tid >> 5;
    const int lm   = lane & 15;       // M (or Ncol) index of this lane
    const int lh   = lane >> 4;       // half-wave selector (K offset)
    const int nb   = blockIdx.x * 64; // block node base

    // ---- stage 64x64 node_features tile into LDS ----
#ifdef HAVE_TDM
    if (wv == 0) {
        // pad_interval=5 -> 1 pad DWORD every 64 DWORDs -> row stride 65
        tdm_load_tile((uint32_t)(uintptr_t)sNF, nf + (long)nb * Hdim,
                      /*row_len=*/64, /*tile_rows=*/64,
                      /*tensor_rows=*/(uint32_t)(N - nb), /*pad_interval=*/5);
        __builtin_amdgcn_s_wait_tensorcnt(0);
    }
#else
    #pragma unroll
    for (int i = 0; i < 32; ++i) {
        int idx = tid + 128 * i;              // 0..4095
        int row = idx >> 6, col = idx & 63;
        float v = 0.f;
        if (nb + row < N) v = nf[(long)(nb + row) * Hdim + col];
        sNF[row * 65 + col] = v;
    }
#endif
    __syncthreads();

    const int myrow = wv * 16 + lm;           // node-in-block for A fragments

    // ---- GEMM1 + ScaledSiLU -> sH ----
    #pragma unroll
    for (int ct = 0; ct < 2; ++ct) {
        v8f acc = {};
        #pragma unroll
        for (int k = 0; k < 16; ++k) {
            const int kk = 4 * k + 2 * lh;
            v2f a, b;
            a.x = sNF[myrow * 65 + kk + 0];
            a.y = sNF[myrow * 65 + kk + 1];
            b.x = Wf1[(kk + 0) * 32 + ct * 16 + lm];
            b.y = Wf1[(kk + 1) * 32 + ct * 16 + lm];
            acc = __builtin_amdgcn_wmma_f32_16x16x4_f32(
                false, a, false, b, (short)0, acc, false, false);
        }
        const int col  = ct * 16 + lm;
        const float bs = bf1[col];
        #pragma unroll
        for (int v = 0; v < 8; ++v) {
            float x = acc[v] + bs;
            float y = x * (1.0f / (1.0f + __expf(-x))) * SILU_SCALEf;
            sH[(wv * 16 + v + 8 * lh) * 33 + col] = y;
        }
    }
    __syncthreads();

    // ---- GEMM2 -> feat_w ----
    #pragma unroll
    for (int ct = 0; ct < 12; ++ct) {
        v8f acc = {};
        #pragma unroll
        for (int k = 0; k < 8; ++k) {
            const int kk = 4 * k + 2 * lh;
            v2f a, b;
            a.x = sH[(wv * 16 + lm) * 33 + kk + 0];
            a.y = sH[(wv * 16 + lm) * 33 + kk + 1];
            b.x = Wf2[(kk + 0) * TH3 + ct * 16 + lm];
            b.y = Wf2[(kk + 1) * TH3 + ct * 16 + lm];
            acc = __builtin_amdgcn_wmma_f32_16x16x4_f32(
                false, a, false, b, (short)0, acc, false, false);
        }
        const int col  = ct * 16 + lm;
        const float bs = bf2[col];
        #pragma unroll
        for (int v = 0; v < 8; ++v) {
            int node = nb + wv * 16 + v + 8 * lh;
            if (node < N) feat_w[(long)node * TH3 + col] = acc[v] + bs;
        }
    }
}

// -------------------------------------------------------------- fused edge kernel
// block = 128 threads (4 waves) -> 64 edges. Per wave: 16 edges.
//   Phase 1: radial_w tile = edge_radial[16x32] @ Wr[32x192] + br via WMMA -> LDS
//   Phase 2: gather feat_w[src], combine with radial_w, atomic scatter-add.
__global__ void __launch_bounds__(128)
emsg_edge_kernel(const float* __restrict__ vecf,   // [N,3,64]
                 const int*   __restrict__ eidx,   // [2,E]
                 const float* __restrict__ erad,   // [E,32]
                 const float* __restrict__ evec,   // [E,3]
                 const float* __restrict__ Wr,     // [32,192]
                 const float* __restrict__ br,     // [192]
                 const float* __restrict__ feat_w, // [N,192]
                 float* __restrict__ out_scal,     // [N,64]
                 float* __restrict__ out_vec,      // [N,3,64]
                 int E) {
    __shared__ float sER[64 * 33];     // 64 edges x 32 radial, stride 33
    __shared__ float sRW[64 * 193];    // 64 edges x 192 combined radial, stride 193
    __shared__ int   sSrc[64];
    __shared__ int   sTgt[64];

    const int tid  = threadIdx.x;
    const int lane = tid & 31;
    const int wv   = tid >> 5;
    const int lm   = lane & 15;
    const int lh   = lane >> 4;
    const int eb   = blockIdx.x * 64;  // block edge base

    // ---- stage edge_radial tile into LDS (TDM DMA, overlaps index loads) ----
#ifdef HAVE_TDM
    if (wv == 0) {
        // pad_interval=4 -> 1 pad DWORD every 32 DWORDs -> row stride 33
        tdm_load_tile((uint32_t)(uintptr_t)sER, erad + (long)eb * Rdim,
                      /*row_len=*/32, /*tile_rows=*/64,
                      /*tensor_rows=*/(uint32_t)(E - eb), /*pad_interval=*/4);
    }
#else
    #pragma unroll
    for (int i = 0; i < 16; ++i) {
        int idx = tid + 128 * i;               // 0..2047
        int row = idx >> 5, col = idx & 31;
        float v = 0.f;
        if (eb + row < E) v = erad[(long)(eb + row) * Rdim + col];
        sER[row * 33 + col] = v;
    }
#endif
    if (tid < 64) {
        int eg = eb + tid;
        sSrc[tid] = (eg < E) ? eidx[eg] : 0;
        sTgt[tid] = (eg < E) ? eidx[(long)E + eg] : 0;
    }
#ifdef HAVE_TDM
    if (wv == 0) {
        __builtin_amdgcn_s_wait_tensorcnt(0);
    }
#endif
    __syncthreads();

    // ---- Phase 1: radial GEMM, 12 column tiles x 8 K-steps of WMMA ----
    #pragma unroll
    for (int ct = 0; ct < 12; ++ct) {
        v8f acc = {};
        #pragma unroll
        for (int k = 0; k < 8; ++k) {
            const int kk = 4 * k + 2 * lh;
            v2f a, b;
            a.x = sER[(wv * 16 + lm) * 33 + kk + 0];
            a.y = sER[(wv * 16 + lm) * 33 + kk + 1];
            b.x = Wr[(kk + 0) * TH3 + ct * 16 + lm];
            b.y = Wr[(kk + 1) * TH3 + ct * 16 + lm];
            acc = __builtin_amdgcn_wmma_f32_16x16x4_f32(
                false, a, false, b, (short)0, acc, false, false);
        }
        const int col  = ct * 16 + lm;
        const float bs = br[col];
        #pragma unroll
        for (int v = 0; v < 8; ++v) {
            sRW[(wv * 16 + v + 8 * lh) * 193 + col] = acc[v] + bs;
        }
    }
    __syncthreads();

    // ---- Phase 2: per-wave scatter for its own 16 edges ----
    for (int e = 0; e < 16; ++e) {
        const int el = wv * 16 + e;
        const int eg = eb + el;
        if (eg >= E) continue;
        const long s = sSrc[el];
        const long t = sTgt[el];
        const float ev0 = evec[(long)eg * 3 + 0];
        const float ev1 = evec[(long)eg * 3 + 1];
        const float ev2 = evec[(long)eg * 3 + 2];
        #pragma unroll
        for (int half = 0; half < 2; ++half) {
            const int h = half * 32 + lane;    // channel 0..63
            const float r1 = sRW[el * 193 + h];
            const float r2 = sRW[el * 193 + 64 + h];
            const float r3 = sRW[el * 193 + 128 + h];
            const float f1 = feat_w[s * TH3 + h];
            const float f2 = feat_w[s * TH3 + 64 + h];
            const float f3 = feat_w[s * TH3 + 128 + h];
            const float w1 = f1 * r1 * INV_SQRT_3f;
            const float w2 = f2 * r2 * INV_SQRT_3f;
            const float w3 = f3 * r3 * INV_SQRT_3f;
            atomicAdd(&out_scal[t * Hdim + h], w3);
            const float vf0 = vecf[s * TH3 + 0 * Hdim + h];
            const float vf1 = vecf[s * TH3 + 1 * Hdim + h];
            const float vf2 = vecf[s * TH3 + 2 * Hdim + h];
            atomicAdd(&out_vec[t * TH3 + 0 * Hdim + h], (w1 * vf0 + w2 * ev0) * INV_SQRT_Hf);
            atomicAdd(&out_vec[t * TH3 + 1 * Hdim + h], (w1 * vf1 + w2 * ev1) * INV_SQRT_Hf);
            atomicAdd(&out_vec[t * TH3 + 2 * Hdim + h], (w1 * vf2 + w2 * ev2) * INV_SQRT_Hf);
        }
    }
}

// ---------------------------------------------------------------------------
extern "C" void kernel_launch(void* const* d_in, const int* in_sizes, int n_in,
                              void* d_out, int out_size, void* d_ws, size_t ws_size,
                              hipStream_t stream) {
    const float* node_features  = (const float*)d_in[0];   // [N,64]
    const float* vector_features= (const float*)d_in[1];   // [N,3,64]
    const int*   edge_index     = (const int*)  d_in[2];   // [2,E]
    const float* edge_radial    = (const float*)d_in[3];   // [E,32]
    const float* edge_vectors   = (const float*)d_in[4];   // [E,3]
    const float* Wf1            = (const float*)d_in[5];   // [64,32]
    const float* bf1            = (const float*)d_in[6];   // [32]
    const float* Wf2            = (const float*)d_in[7];   // [32,192]
    const float* bf2            = (const float*)d_in[8];   // [192]
    const float* Wr             = (const float*)d_in[9];   // [32,192]
    const float* br             = (const float*)d_in[10];  // [192]

    const int N = in_sizes[0] / Hdim;
    const int E = in_sizes[3] / Rdim;

    float* out_scal = (float*)d_out;                 // [N,64]
    float* out_vec  = (float*)d_out + (long)N * Hdim;// [N,3,64]
    float* feat_w   = (float*)d_ws;                  // [N,192] scratch

    // 1) zero accumulation targets (atomics accumulate into them)
    {
        long n4 = (long)out_size / 4;  // out_size = N*256, divisible by 4
        long grid = (n4 + 255) / 256;
        emsg_zero_kernel<<<(int)grid, 256, 0, stream>>>((float*)d_out, n4);
    }
    // 2) node feature projector -> feat_w (workspace)
    {
        int grid = (N + 63) / 64;
        emsg_featproj_kernel<<<grid, 128, 0, stream>>>(
            node_features, Wf1, bf1, Wf2, bf2, feat_w, N);
    }
    // 3) fused radial GEMM + gather + scatter
    {
        int grid = (E + 63) / 64;
        emsg_edge_kernel<<<grid, 128, 0, stream>>>(
            vector_features, edge_index, edge_radial, edge_vectors,
            Wr, br, feat_w, out_scal, out_vec, E);
    }
}